// PhaserSampleBased_19155554140785
// MI455X (gfx1250) — compile-verified
//
#include <hip/hip_runtime.h>
#include <math.h>

// ---------------- problem constants (from reference) ----------------
#define T_LEN     (1 << 20)           // samples
#define HOPX      512
#define H_FRAMES  ((T_LEN / HOPX) + 2) // 2050 hop frames
#define POLY_M    7                    // coefficient length after K=4 polymuls
#define CB_STRIDE 8                    // padded row stride for coeff tables

// chunked-recurrence geometry
#define L1 512
#define C1 (T_LEN / L1)   // 2048 chunks, order-2 stage
#define L2 512
#define C2 (T_LEN / L2)   // 2048 chunks, order-6 stage

typedef __attribute__((ext_vector_type(2))) float v2f;
typedef __attribute__((ext_vector_type(8))) float v8f;

// =====================================================================
// Kernel 1: LFO + MLP (WMMA f32 16x16x4 for the two 16x16 layers)
//           + per-frame allpass coefficient construction.
// One wave (32 threads) per 16-row frame tile; grid = ceil(2050/16)=129.
// =====================================================================
__global__ void mlp_coeff_kernel(
    const float* __restrict__ g2,
    const float* __restrict__ depth,
    const float* __restrict__ biasP,
    const float* __restrict__ omega,
    const float* __restrict__ rP,
    const float* __restrict__ z0,
    const float* __restrict__ W0, const float* __restrict__ b0,
    const float* __restrict__ W1, const float* __restrict__ b1,
    const float* __restrict__ W2, const float* __restrict__ b2,
    const float* __restrict__ W3, const float* __restrict__ b3,
    const float* __restrict__ f2b, const float* __restrict__ f2a,
    float* __restrict__ cbT, float* __restrict__ daT)
{
    __shared__ float hs[16][17];       // 17 stride: bank-conflict-free cols
    const int lane = threadIdx.x;      // wave32
    const int half = lane >> 4;
    const int l15  = lane & 15;
    const int m0   = blockIdx.x * 16;

    // ---- layer 0 (1 -> 16): lanes 0..15 each own one frame row ----
    if (lane < 16) {
        float t  = (float)(m0 + lane);
        float r  = rP[0], om = omega[0];
        float lfo = powf(r, t) * (z0[0] * cosf(om * t) - z0[1] * sinf(om * t));
        #pragma unroll
        for (int j = 0; j < 16; ++j)
            hs[lane][j] = tanhf(lfo * W0[j] + b0[j]);
    }
    __syncthreads();

    // ---- layers 1,2 (16 -> 16): D = tanh(A x W^T + b) via WMMA ----
    const float* Ws[2] = { W1, W2 };
    const float* bs[2] = { b1, b2 };
    #pragma unroll
    for (int lay = 0; lay < 2; ++lay) {
        const float* Wl = Ws[lay];
        const float* bl = bs[lay];

        // B tiles (4x16 f32 per chunk): B[k][n] = Wl[n*16 + k]
        // lanes 0..15 hold K = k0, k0+1 ; lanes 16..31 hold K = k0+2, k0+3
        v2f btile[4];
        #pragma unroll
        for (int kk = 0; kk < 4; ++kk) {
            int k0 = kk * 4 + half * 2;
            btile[kk].x = Wl[l15 * 16 + k0];
            btile[kk].y = Wl[l15 * 16 + k0 + 1];
        }

        // C init = bias broadcast along rows
        float bias_n = bl[l15];
        v8f c;
        #pragma unroll
        for (int rI = 0; rI < 8; ++rI) c[rI] = bias_n;

        // K = 16 as 4 x (16x16x4) WMMAs; EXEC is all-ones here (full wave)
        #pragma unroll
        for (int kk = 0; kk < 4; ++kk) {
            int k0 = kk * 4 + half * 2;
            v2f a;                      // A 16x4 f32: M = l15, K = k0,k0+1
            a.x = hs[l15][k0];
            a.y = hs[l15][k0 + 1];
            c = __builtin_amdgcn_wmma_f32_16x16x4_f32(
                    false, a, false, btile[kk], (short)0, c, false, false);
        }
        __syncthreads();
        // C/D layout: VGPR r -> row (r + half*8), col l15
        #pragma unroll
        for (int rI = 0; rI < 8; ++rI)
            hs[rI + half * 8][l15] = tanhf(c[rI]);
        __syncthreads();
    }

    // ---- layer 3 (16 -> 1) + allpass coefficient chain ----
    if (lane < 16) {
        int t = m0 + lane;
        if (t < H_FRAMES) {
            float acc = b3[0];
            #pragma unroll
            for (int j = 0; j < 16; ++j) acc += hs[lane][j] * W3[j];
            float wsv = tanhf(acc);

            float d  = biasP[0] + depth[0] * 0.5f * (1.0f + wsv);
            float td = tanf(d);
            float p  = (1.0f - td) / (1.0f + td);

            float cb[POLY_M] = {1.f, f2b[0], f2b[1], 0.f, 0.f, 0.f, 0.f};
            float ca[POLY_M] = {1.f, f2a[0], f2a[1], 0.f, 0.f, 0.f, 0.f};
            #pragma unroll
            for (int it = 0; it < 4; ++it) {     // K = 4 polymuls
                #pragma unroll
                for (int i = POLY_M - 1; i >= 0; --i) {
                    float cbm1 = (i > 0) ? cb[i - 1] : 0.f;
                    float cam1 = (i > 0) ? ca[i - 1] : 0.f;
                    cb[i] = cb[i] * p - cbm1;    // * (p, -1)
                    ca[i] = ca[i] - p * cam1;    // * (1, -p)
                }
            }
            float ag2 = fabsf(g2[0]);
            float inv = 1.0f / (ca[0] - ag2 * cb[0]);
            #pragma unroll
            for (int i = 0; i < POLY_M; ++i) {
                float deni = ca[i] - ag2 * cb[i];
                cbT[t * CB_STRIDE + i] = cb[i] * inv;
                daT[t * CB_STRIDE + i] = deni * inv;
            }
        }
    }
}

// =====================================================================
// Kernel 2: 3-tap FIR   fir1[n] = DC*x[n] + b0*x[n-1] + b1*x[n-2]
// =====================================================================
__global__ void fir1_kernel(const float* __restrict__ x,
                            const float* __restrict__ dc,
                            const float* __restrict__ f1b,
                            float* __restrict__ out)
{
    int n = blockIdx.x * blockDim.x + threadIdx.x;
    if (n >= T_LEN) return;
    float v = dc[0] * x[n];
    if (n >= 1) v += f1b[0] * x[n - 1];
    if (n >= 2) v += f1b[1] * x[n - 2];
    out[n] = v;
}

// =====================================================================
// Chunked time-varying IIR:  y[n] = in[n] - sum_k a_k[n] * y[n-k]
// Phase A: per (chunk, j) end-states (j=0 particular, j=1..P unit states)
// Phase B: serial cross-chunk state propagation (tiny)
// Phase C: corrected per-chunk re-run, writes full output
// TV=true reads interpolated coefficients from table (stride CB_STRIDE,
// columns 1..P), TV=false uses constant coefficients aconst[0..P-1].
// =====================================================================
template <int P, bool TV>
__global__ void iirA_kernel(const float* __restrict__ in,
                            const float* __restrict__ aconst,
                            const float* __restrict__ coef,
                            int L, int C,
                            float* __restrict__ endst)
{
    int tid = blockIdx.x * blockDim.x + threadIdx.x;
    if (tid >= C * (P + 1)) return;
    int c = tid / (P + 1);
    int j = tid - c * (P + 1);

    float st[P];
    #pragma unroll
    for (int k = 0; k < P; ++k) st[k] = 0.f;
    if (j > 0) st[j - 1] = 1.f;

    float ac[P];
    if (!TV) {
        #pragma unroll
        for (int k = 0; k < P; ++k) ac[k] = aconst[k];
    }

    int base = c * L;
    for (int n = 0; n < L; ++n) {
        int idx = base + n;
        __builtin_prefetch(in + idx + 256, 0, 0);   // global_prefetch_b8
        float yn = (j == 0) ? in[idx] : 0.f;
        if (TV) {
            int   i0   = idx >> 9;                        // / HOP
            float frac = (float)(idx & 511) * (1.f / 512.f);
            #pragma unroll
            for (int k = 0; k < P; ++k) {
                float c0 = coef[i0 * CB_STRIDE + k + 1];
                float c1 = coef[(i0 + 1) * CB_STRIDE + k + 1];
                ac[k] = c0 + (c1 - c0) * frac;
            }
        }
        #pragma unroll
        for (int k = 0; k < P; ++k) yn -= ac[k] * st[k];
        #pragma unroll
        for (int i = P - 1; i > 0; --i) st[i] = st[i - 1];
        st[0] = yn;
    }
    #pragma unroll
    for (int k = 0; k < P; ++k)
        endst[(c * (P + 1) + j) * P + k] = st[k];
}

template <int P>
__global__ void iirB_kernel(const float* __restrict__ endst,
                            int C, float* __restrict__ init)
{
    if (threadIdx.x != 0 || blockIdx.x != 0) return;
    float s[P];
    #pragma unroll
    for (int k = 0; k < P; ++k) s[k] = 0.f;   // zero initial conditions
    for (int c = 0; c < C; ++c) {
        #pragma unroll
        for (int k = 0; k < P; ++k) init[c * P + k] = s[k];
        float ns[P];
        #pragma unroll
        for (int k = 0; k < P; ++k) {
            float v = endst[(c * (P + 1) + 0) * P + k];   // particular
            #pragma unroll
            for (int j = 1; j <= P; ++j)                  // homogeneous
                v += endst[(c * (P + 1) + j) * P + k] * s[j - 1];
            ns[k] = v;
        }
        #pragma unroll
        for (int k = 0; k < P; ++k) s[k] = ns[k];
    }
}

template <int P, bool TV>
__global__ void iirC_kernel(const float* __restrict__ in,
                            const float* __restrict__ aconst,
                            const float* __restrict__ coef,
                            int L, int C,
                            const float* __restrict__ init,
                            float* __restrict__ out)
{
    int c = blockIdx.x * blockDim.x + threadIdx.x;
    if (c >= C) return;

    float st[P];
    #pragma unroll
    for (int k = 0; k < P; ++k) st[k] = init[c * P + k];

    float ac[P];
    if (!TV) {
        #pragma unroll
        for (int k = 0; k < P; ++k) ac[k] = aconst[k];
    }

    int base = c * L;
    for (int n = 0; n < L; ++n) {
        int idx = base + n;
        __builtin_prefetch(in + idx + 256, 0, 0);
        float yn = in[idx];
        if (TV) {
            int   i0   = idx >> 9;
            float frac = (float)(idx & 511) * (1.f / 512.f);
            #pragma unroll
            for (int k = 0; k < P; ++k) {
                float c0 = coef[i0 * CB_STRIDE + k + 1];
                float c1 = coef[(i0 + 1) * CB_STRIDE + k + 1];
                ac[k] = c0 + (c1 - c0) * frac;
            }
        }
        #pragma unroll
        for (int k = 0; k < P; ++k) yn -= ac[k] * st[k];
        #pragma unroll
        for (int i = P - 1; i > 0; --i) st[i] = st[i - 1];
        st[0] = yn;
        out[idx] = yn;
    }
}

// =====================================================================
// Kernel: time-varying 7-tap FIR with interpolated numerator coeffs
// =====================================================================
__global__ void fir2_kernel(const float* __restrict__ h1,
                            const float* __restrict__ cbT,
                            float* __restrict__ out)
{
    int n = blockIdx.x * blockDim.x + threadIdx.x;
    if (n >= T_LEN) return;
    int   i0   = n >> 9;
    float frac = (float)(n & 511) * (1.f / 512.f);
    float acc = 0.f;
    #pragma unroll
    for (int k = 0; k < POLY_M; ++k) {
        float c0 = cbT[i0 * CB_STRIDE + k];
        float c1 = cbT[(i0 + 1) * CB_STRIDE + k];
        float ck = c0 + (c1 - c0) * frac;
        float xv = (n >= k) ? h1[n - k] : 0.f;
        acc += xv * ck;
    }
    out[n] = acc;
}

__global__ void out_kernel(const float* __restrict__ h1,
                           const float* __restrict__ y,
                           const float* __restrict__ g1,
                           float* __restrict__ out)
{
    int n = blockIdx.x * blockDim.x + threadIdx.x;
    if (n >= T_LEN) return;
    out[n] = g1[0] * h1[n] + y[n];
}

// =====================================================================
// Launcher
// =====================================================================
extern "C" void kernel_launch(void* const* d_in, const int* in_sizes, int n_in,
                              void* d_out, int out_size, void* d_ws, size_t ws_size,
                              hipStream_t stream)
{
    const float* x     = (const float*)d_in[0];
    const float* g1    = (const float*)d_in[1];
    const float* g2    = (const float*)d_in[2];
    const float* depth = (const float*)d_in[3];
    const float* biasP = (const float*)d_in[4];
    const float* omega = (const float*)d_in[5];
    const float* rP    = (const float*)d_in[6];
    const float* z0    = (const float*)d_in[7];
    const float* W0    = (const float*)d_in[8];
    const float* b0    = (const float*)d_in[9];
    const float* W1    = (const float*)d_in[10];
    const float* b1    = (const float*)d_in[11];
    const float* W2    = (const float*)d_in[12];
    const float* b2    = (const float*)d_in[13];
    const float* W3    = (const float*)d_in[14];
    const float* b3    = (const float*)d_in[15];
    const float* f1b   = (const float*)d_in[16];
    const float* f1a   = (const float*)d_in[17];
    const float* f1dc  = (const float*)d_in[18];
    const float* f2b   = (const float*)d_in[19];
    const float* f2a   = (const float*)d_in[20];
    float* out = (float*)d_out;

    // workspace layout (floats) — ~17.4 MB total
    float* ws    = (float*)d_ws;
    float* cbT   = ws;                        // 2064*8
    float* daT   = cbT   + 2064 * CB_STRIDE;  // 2064*8
    float* fir1  = daT   + 2064 * CB_STRIDE;  // T
    float* h1    = fir1  + T_LEN;             // T
    float* fir2  = h1    + T_LEN;             // T
    float* ybuf  = fir2  + T_LEN;             // T
    float* endA1 = ybuf  + T_LEN;             // C1*3*2
    float* init1 = endA1 + C1 * 3 * 2;        // C1*2
    float* endA2 = init1 + C1 * 2;            // C2*7*6
    float* init2 = endA2 + C2 * 7 * 6;        // C2*6

    // control path: LFO + MLP (WMMA) + coefficient tables
    mlp_coeff_kernel<<<(H_FRAMES + 15) / 16, 32, 0, stream>>>(
        g2, depth, biasP, omega, rP, z0,
        W0, b0, W1, b1, W2, b2, W3, b3, f2b, f2a, cbT, daT);

    // audio path
    fir1_kernel<<<T_LEN / 256, 256, 0, stream>>>(x, f1dc, f1b, fir1);

    iirA_kernel<2, false><<<(C1 * 3 + 127) / 128, 128, 0, stream>>>(
        fir1, f1a, nullptr, L1, C1, endA1);
    iirB_kernel<2><<<1, 1, 0, stream>>>(endA1, C1, init1);
    iirC_kernel<2, false><<<(C1 + 127) / 128, 128, 0, stream>>>(
        fir1, f1a, nullptr, L1, C1, init1, h1);

    fir2_kernel<<<T_LEN / 256, 256, 0, stream>>>(h1, cbT, fir2);

    iirA_kernel<6, true><<<(C2 * 7 + 127) / 128, 128, 0, stream>>>(
        fir2, nullptr, daT, L2, C2, endA2);
    iirB_kernel<6><<<1, 1, 0, stream>>>(endA2, C2, init2);
    iirC_kernel<6, true><<<(C2 + 127) / 128, 128, 0, stream>>>(
        fir2, nullptr, daT, L2, C2, init2, ybuf);

    out_kernel<<<T_LEN / 256, 256, 0, stream>>>(h1, ybuf, g1, out);
}